// FPSKNNGroup_12781822673371
// MI455X (gfx1250) — compile-verified
//
#include <hip/hip_runtime.h>
#include <float.h>

// ---------------- static problem config (matches reference) ----------------
constexpr int Bn  = 16;      // batches
constexpr int NB  = 8192;    // points per batch
constexpr int Mn  = 1024;    // FPS samples per batch
constexpr int Kn  = 16;      // neighbors
constexpr int T1  = 256;     // threads, fps kernel
constexpr int PPT = NB / T1; // 32 points per thread (register resident)
constexpr int TM  = 16;      // centroid rows per knn block
constexpr int NIT = NB / 128; // 64 tile-steps in knn (8 waves x 16 pts)

typedef __attribute__((ext_vector_type(2))) float v2f;
typedef __attribute__((ext_vector_type(8))) float v8f;

// ---------------------------------------------------------------------
// Workgroup barrier that drains ONLY the LDS counter before signaling.
// __syncthreads() lowers to s_wait_loadcnt_dscnt 0x0 (+ storecnt), which
// would force in-flight global prefetch loads / centroid stores to
// complete on the barrier critical path. Only DScnt is semantically
// required here: each wave's LDS stores must be visible before it
// signals. Global loads/stores are waited at their true use sites by
// the compiler's waitcnt insertion (loads) or kernel-end (stores).
// ---------------------------------------------------------------------
__device__ __forceinline__ void barrier_ds_only()
{
    asm volatile(
        "s_wait_dscnt 0x0\n"
        "s_barrier_signal -1\n"
        "s_barrier_wait -1"
        ::: "memory");
}

// =====================================================================
// Kernel 1: farthest point sampling. One block per batch.
// Distance state and each thread's 32 points live in registers; a full
// coordinate mirror sits in LDS (96KB of the WGP's 320KB) purely so the
// argmax winner's coords can be read by broadcast ds_load instead of a
// second barrier + owner republish. Cross-wave argmax keys are parity
// double-buffered -> exactly ONE ds-only barrier per sequential step,
// and thread0's global centroid store never stalls the chain.
// =====================================================================
__global__ __launch_bounds__(T1) void fps_kernel(const float* __restrict__ pos,
                                                 float* __restrict__ centroids)
{
    const int b    = blockIdx.x;
    const int tid  = threadIdx.x;
    const int lane = tid & 31;
    const int wave = tid >> 5;

    __shared__ unsigned long long skey[2][8];   // parity double-buffered
    extern __shared__ float lp[];               // [3*NB]: x-plane | y | z
    float* lx = lp;
    float* ly = lp + NB;
    float* lz = lp + 2 * NB;

    const float* pb = pos + (size_t)b * NB * 3;

    float px[PPT], py[PPT], pz[PPT], md[PPT];
#pragma unroll
    for (int j = 0; j < PPT; ++j) {
        const int n = tid + j * T1;
        const float x = pb[n * 3 + 0];
        const float y = pb[n * 3 + 1];
        const float z = pb[n * 3 + 2];
        px[j] = x; py[j] = y; pz[j] = z; md[j] = FLT_MAX;
        lx[n] = x; ly[n] = y; lz[n] = z;   // LDS mirror (stride-1 planes)
    }
    barrier_ds_only();

    float* outB = centroids + (size_t)b * Mn * 3;
    int cur = 0;                                // deterministic start at 0

    for (int i = 0; i < Mn; ++i) {
        // broadcast read of current farthest point (conflict-free: 1 addr)
        const float cx = lx[cur], cy = ly[cur], cz = lz[cur];
        if (tid == 0) {
            outB[i * 3 + 0] = cx; outB[i * 3 + 1] = cy; outB[i * 3 + 2] = cz;
        }
        if (i == Mn - 1) break;

        // fused min-distance update + local argmax over updated values
        float bd = -1.0f; int bi = 0;
#pragma unroll
        for (int j = 0; j < PPT; ++j) {
            const float dx = px[j] - cx, dy = py[j] - cy, dz = pz[j] - cz;
            const float d2 = dx * dx + dy * dy + dz * dz;
            const float m  = fminf(md[j], d2);
            md[j] = m;
            if (m > bd) { bd = m; bi = tid + j * T1; }
        }
        // pack (dist_bits, ~idx): max -> largest dist, smallest idx on ties
        unsigned long long key =
            ((unsigned long long)__float_as_uint(bd) << 32) |
            (unsigned)(0xFFFFFFFFu - (unsigned)bi);
#pragma unroll
        for (int off = 16; off > 0; off >>= 1) {
            const unsigned long long o = __shfl_xor(key, off, 32);
            key = (o > key) ? o : key;
        }
        if (lane == 0) skey[i & 1][wave] = key;
        barrier_ds_only();                      // the ONLY barrier per step

        unsigned long long kb = skey[i & 1][0];
#pragma unroll
        for (int w = 1; w < 8; ++w) {
            const unsigned long long o = skey[i & 1][w];
            kb = (o > kb) ? o : kb;
        }
        cur = (int)(0xFFFFFFFFu - (unsigned)kb);
    }
}

// =====================================================================
// Kernel 2: kNN via V_WMMA_F32_16X16X4_F32 + per-thread top-16 + merge.
// One block = (batch, 16-centroid tile); 8 waves x 64 steps cover 8192 pts.
// Score tile D[m][n] = ||p_n||^2 - 2 c_m . p_n  (row-constant ||c||^2
// dropped: does not change per-row top-K ranking).
//   A row m = [-2cx, -2cy, -2cz, 1]      (16x4 f32)
//   B col n = [ px,   py,   pz, ||p||^2] (4x16 f32)
// VGPR layouts per CDNA5 ISA 7.12.2 (B assumed to mirror A's K-halving).
// Per step: 1 ds-only barrier (d2s tiles parity double-buffered); the
// next step's point coords are prefetched BEFORE the barrier and now
// genuinely stay in flight across barrier+selection.
// =====================================================================
__global__ __launch_bounds__(256) void knn_kernel(const float* __restrict__ pos,
                                                  const float* __restrict__ centroids,
                                                  float* __restrict__ groups)
{
    const int blk  = blockIdx.x;      // 0 .. B*(M/16)-1
    const int b    = blk >> 6;        // / 64
    const int mt   = blk & 63;
    const int tid  = threadIdx.x;
    const int lane = tid & 31;
    const int wave = tid >> 5;
    const int half = lane >> 4;
    const int l16  = lane & 15;

    __shared__ float d2s[2][8][TM][16];    // parity double-buffered tiles
    __shared__ float candd[TM][16][Kn];    // per-(row,slice) sorted top-16
    __shared__ int   candi[TM][16][Kn];
    __shared__ int   nnIdx[TM][Kn];
    __shared__ int   head[TM][16];

    const float* pb = pos + (size_t)b * NB * 3;
    const float* cb = centroids + ((size_t)b * Mn + (size_t)mt * TM) * 3;

    // ---- A fragment (loop invariant) ----
    const float cx = cb[l16 * 3 + 0];
    const float cy = cb[l16 * 3 + 1];
    const float cz = cb[l16 * 3 + 2];
    v2f afrag;
    afrag.x = half ? (-2.0f * cz) : (-2.0f * cx);   // K2 : K0
    afrag.y = half ? 1.0f         : (-2.0f * cy);   // K3 : K1

    // ---- per-thread selection state: row m, slice s ----
    const int m  = tid & 15;
    const int s  = tid >> 4;        // 0..15
    const int sw = s & 7;           // source wave tile
    const int cg = (s >> 3) * 8;    // column group (0 or 8)
    float hd[Kn]; int hi[Kn];
#pragma unroll
    for (int k = 0; k < Kn; ++k) { hd[k] = FLT_MAX; hi[k] = 0; }

    // prologue: load step-0 point coords
    int nq = wave * 16 + l16;
    float qx = pb[nq * 3 + 0];
    float qy = pb[nq * 3 + 1];
    float qz = pb[nq * 3 + 2];

    for (int it = 0; it < NIT; ++it) {
        // -------- score tile via WMMA (no divergence: EXEC all ones) --------
        v2f bfrag;
        bfrag.x = half ? qz : qx;                              // K2 : K0
        bfrag.y = half ? (qx * qx + qy * qy + qz * qz) : qy;   // K3 : K1

        v8f acc = {};
        acc = __builtin_amdgcn_wmma_f32_16x16x4_f32(
            /*neg_a=*/false, afrag, /*neg_b=*/false, bfrag,
            /*c_mod=*/(short)0, acc, /*reuse_a=*/false, /*reuse_b=*/false);

        const int buf = it & 1;
#pragma unroll
        for (int v = 0; v < 8; ++v)
            d2s[buf][wave][v + 8 * half][l16] = acc[v];

        // -------- prefetch next step's point over barrier+selection --------
        const int itn = (it + 1 < NIT) ? (it + 1) : it;   // branchless clamp
        nq = (itn * 8 + wave) * 16 + l16;
        qx = pb[nq * 3 + 0];
        qy = pb[nq * 3 + 1];
        qz = pb[nq * 3 + 2];

        barrier_ds_only();             // drains DScnt only; prefetch flies over

        // -------- streaming top-16 insertion (8 candidates, 2x b128) -------
        const int nb2 = (it * 8 + sw) * 16 + cg;
        const float4* row = reinterpret_cast<const float4*>(&d2s[buf][sw][m][cg]);
        const float4 r0 = row[0];
        const float4 r1 = row[1];
        const float dv[8] = { r0.x, r0.y, r0.z, r0.w, r1.x, r1.y, r1.z, r1.w };
#pragma unroll
        for (int c = 0; c < 8; ++c) {
            const float d = dv[c];
            if (d < hd[Kn - 1]) {                 // usually skipped
                hd[Kn - 1] = d; hi[Kn - 1] = nb2 + c;
#pragma unroll
                for (int j = Kn - 1; j > 0; --j) {
                    if (hd[j] < hd[j - 1]) {
                        const float tf = hd[j]; hd[j] = hd[j - 1]; hd[j - 1] = tf;
                        const int   ti = hi[j]; hi[j] = hi[j - 1]; hi[j - 1] = ti;
                    }
                }
            }
        }
        // no trailing barrier: next step writes the other d2s parity buffer
    }

    // ---- dump sorted lists, init merge heads ----
#pragma unroll
    for (int k = 0; k < Kn; ++k) { candd[m][s][k] = hd[k]; candi[m][s][k] = hi[k]; }
    head[m][s] = 0;
    barrier_ds_only();

    // ---- 16-way merge, one thread per centroid row ----
    if (tid < TM) {
        const int mm = tid;
        for (int k = 0; k < Kn; ++k) {
            float best = FLT_MAX; int bs = 0;
            for (int ss = 0; ss < 16; ++ss) {
                const int h = head[mm][ss];
                const float v = (h < Kn) ? candd[mm][ss][h] : FLT_MAX;
                if (v < best) { best = v; bs = ss; }
            }
            nnIdx[mm][k] = candi[mm][bs][head[mm][bs]];
            head[mm][bs] += 1;
        }
    }
    barrier_ds_only();

    // ---- parallel gather: one (row, k) pair per thread ----
    {
        const int mm = tid >> 4, kk = tid & 15;
        const int nn = nnIdx[mm][kk];
        const float* q = pb + (size_t)nn * 3;
        const size_t o = (((size_t)b * Mn + (size_t)mt * TM + mm) * Kn + kk) * 3;
        groups[o + 0] = q[0];
        groups[o + 1] = q[1];
        groups[o + 2] = q[2];
    }
}

// =====================================================================
extern "C" void kernel_launch(void* const* d_in, const int* in_sizes, int n_in,
                              void* d_out, int out_size, void* d_ws, size_t ws_size,
                              hipStream_t stream)
{
    (void)in_sizes; (void)n_in; (void)out_size; (void)d_ws; (void)ws_size;
    // inputs: x (features, not needed for outputs), pos, batch (block-constant)
    const float* pos = (const float*)d_in[1];

    float* centroids = (float*)d_out;                       // B*M*3 floats
    float* groups    = centroids + (size_t)Bn * Mn * 3;     // B*M*K*3 floats

    // 96KB dynamic LDS coordinate mirror (CDNA5 WGP has 320KB)
    const size_t fpsLds = (size_t)3 * NB * sizeof(float);
    fps_kernel<<<Bn, T1, fpsLds, stream>>>(pos, centroids);
    knn_kernel<<<Bn * (Mn / TM), 256, 0, stream>>>(pos, centroids, groups);
}